// RGRec_50148038148223
// MI455X (gfx1250) — compile-verified
//
#include <hip/hip_runtime.h>
#include <math.h>

#define E_NUM 100000
#define DIM 32
#define BATCH 1024
#define NEIGH 16
#define RULES 8

typedef __attribute__((ext_vector_type(2))) float v2f;
typedef __attribute__((ext_vector_type(8))) float v8f;

__device__ __forceinline__ float wave_sum(float v) {
#pragma unroll
  for (int off = 16; off > 0; off >>= 1) v += __shfl_xor(v, off, 32);
  return v;
}

// scale s.t. v*scale == v * min(1, 1/max(||v||, 1e-7))
__device__ __forceinline__ float renorm_scale(float sumsq) {
  float n = sqrtf(sumsq);
  return fminf(1.0f, 1.0f / fmaxf(n, 1e-7f));
}

__launch_bounds__(256)
__global__ void rgrec_kernel(const int* __restrict__ users,
                             const int* __restrict__ items,
                             const int* __restrict__ rules,
                             const int* __restrict__ adj,
                             const float* __restrict__ ent,
                             const float* __restrict__ rule_w,
                             const float* __restrict__ W,
                             const float* __restrict__ bias,
                             float* __restrict__ out) {
  // LDS staging (~52 KB / workgroup; WGP has 320 KB)
  __shared__ float sH1[128 * 64];  // hop1 rows: [renorm self (32) | mean renorm hop2 (32)]
  __shared__ float sH0[16 * 64];   // hop0 tile (rows 8..15 zero pad)
  __shared__ float sWt[64 * 32];   // W transposed: sWt[d*32+o] = W[o*64+d]
  __shared__ float sH2[16 * 64];   // layer-2 input tile
  __shared__ float sRes[16 * 32];  // layer-2 output (tanh)
  __shared__ float sB[32];
  __shared__ int   sN1[128];
  __shared__ int   sRid1[8];

  const int b    = blockIdx.x;
  const int tid  = threadIdx.x;
  const int wave = tid >> 5;
  const int lane = tid & 31;
  const int hi   = lane >> 4;   // which K-half this lane holds in WMMA frags
  const int lm   = lane & 15;   // M (A) / N (B,D) index within tile

  const int user = users[b];

  // ---- Phase 1: hop-1 neighbor ids, rule ids, weights, zero pads ----
  if (tid < 128) {
    int r = tid >> 4, j = tid & 15;
    int rid0 = rules[r * 2 + 0];
    sN1[tid] = adj[(size_t)rid0 * E_NUM * NEIGH + (size_t)user * NEIGH + j];
  }
  if (tid < 8)  sRid1[tid] = rules[tid * 2 + 1];
  if (tid < 32) sB[tid] = bias[tid];
  for (int i = tid; i < 64 * 32; i += 256) {
    int d = i >> 5, o = i & 31;
    sWt[i] = W[o * 64 + d];
  }
  for (int i = tid; i < 16 * 64; i += 256) sH2[i] = 0.0f;
  for (int i = tid; i < 8 * 64; i += 256) sH0[8 * 64 + i] = 0.0f;
  __syncthreads();

  // ---- Phase 2: gather + renorm hop1 self rows and hop2 neighbor means ----
  // wave w owns hop1 rows [w*16, w*16+16); lane = embedding dim
  for (int p = 0; p < 16; ++p) {
    int m = wave * 16 + p;
    int e1 = sN1[m];
    int rid1 = sRid1[m >> 4];
    float x = ent[(size_t)e1 * DIM + lane];
    float sc = renorm_scale(wave_sum(x * x));
    sH1[m * 64 + lane] = x * sc;
    const int* nb = adj + (size_t)rid1 * E_NUM * NEIGH + (size_t)e1 * NEIGH;
    float acc = 0.0f;
#pragma unroll 4
    for (int k = 0; k < NEIGH; ++k) {
      int e2 = nb[k];
      float y = ent[(size_t)e2 * DIM + lane];
      acc += y * renorm_scale(wave_sum(y * y));
    }
    sH1[m * 64 + 32 + lane] = acc * (1.0f / 16.0f);
  }
  __syncthreads();

  // ---- Phase 3: hop0 rows (self = renorm user emb; neigh = mean of 16 hop1 rows) ----
  {
    int r = wave;  // 8 waves -> 8 rule rows
    float xu = ent[(size_t)user * DIM + lane];
    float sc = renorm_scale(wave_sum(xu * xu));
    sH0[r * 64 + lane] = xu * sc;
    float s = 0.0f;
#pragma unroll
    for (int j = 0; j < 16; ++j) s += sH1[(r * 16 + j) * 64 + lane];
    sH0[r * 64 + 32 + lane] = s * (1.0f / 16.0f);
  }
  __syncthreads();

  // ---- Phase 4: layer-1 projection (relu) via V_WMMA_F32_16X16X4_F32 ----
  // jobs 0..15: hop1 tile t=job/2, N-half=job%2 -> reduce rows to neighbor mean
  // jobs 16..17: hop0 tile -> self part of layer-2 input
  for (int job = wave; job < 18; job += 8) {
    const float* A;
    int nHalf, t = 0;
    if (job < 16) { t = job >> 1; nHalf = job & 1; A = &sH1[t * 16 * 64]; }
    else          { nHalf = job - 16;              A = sH0; }
    v8f c = {};
#pragma unroll
    for (int kk = 0; kk < 16; ++kk) {
      int kb = kk * 4 + 2 * hi;
      v2f a, bb;
      a.x  = A[lm * 64 + kb];
      a.y  = A[lm * 64 + kb + 1];
      bb.x = sWt[kb * 32 + nHalf * 16 + lm];
      bb.y = sWt[(kb + 1) * 32 + nHalf * 16 + lm];
      c = __builtin_amdgcn_wmma_f32_16x16x4_f32(false, a, false, bb,
                                                (short)0, c, false, false);
    }
    int o = nHalf * 16 + lm;
    float bv = sB[o];
    if (job < 16) {
      // relu then mean over the 16 rows of this tile (= neighbors of parent t)
      float part = 0.0f;
#pragma unroll
      for (int v = 0; v < 8; ++v) part += fmaxf(c[v] + bv, 0.0f);
      float tot = part + __shfl_xor(part, 16, 32);
      if (lane < 16) sH2[t * 64 + 32 + o] = tot * (1.0f / 16.0f);
    } else {
      if (lane < 16) {  // rows 0..7 valid (lane<16 holds M=0..7)
#pragma unroll
        for (int v = 0; v < 8; ++v) sH2[v * 64 + o] = fmaxf(c[v] + bv, 0.0f);
      }
    }
  }
  __syncthreads();

  // ---- Phase 5: layer-2 projection (tanh) ----
  if (wave < 2) {
    int nHalf = wave;
    v8f c = {};
#pragma unroll
    for (int kk = 0; kk < 16; ++kk) {
      int kb = kk * 4 + 2 * hi;
      v2f a, bb;
      a.x  = sH2[lm * 64 + kb];
      a.y  = sH2[lm * 64 + kb + 1];
      bb.x = sWt[kb * 32 + nHalf * 16 + lm];
      bb.y = sWt[(kb + 1) * 32 + nHalf * 16 + lm];
      c = __builtin_amdgcn_wmma_f32_16x16x4_f32(false, a, false, bb,
                                                (short)0, c, false, false);
    }
    int o = nHalf * 16 + lm;
    float bv = sB[o];
#pragma unroll
    for (int v = 0; v < 8; ++v) {
      int row = v + 8 * hi;
      sRes[row * 32 + o] = tanhf(c[v] + bv);
    }
  }
  __syncthreads();

  // ---- Phase 6: rule-weighted sum, item dot, sigmoid ----
  if (wave == 0) {
    float rw2 = 0.0f;
#pragma unroll
    for (int r = 0; r < RULES; ++r) { float w = rule_w[r]; rw2 += w * w; }
    float rsc = renorm_scale(rw2);
    float uo = 0.0f;
#pragma unroll
    for (int r = 0; r < RULES; ++r) uo += sRes[r * 32 + lane] * (rule_w[r] * rsc);
    int it = items[b];
    float xi = ent[(size_t)it * DIM + lane];
    float isc = renorm_scale(wave_sum(xi * xi));
    float dot = wave_sum(uo * xi * isc);
    if (lane == 0) out[b] = 1.0f / (1.0f + expf(-dot));
  }
}

extern "C" void kernel_launch(void* const* d_in, const int* in_sizes, int n_in,
                              void* d_out, int out_size, void* d_ws, size_t ws_size,
                              hipStream_t stream) {
  const int*   users = (const int*)d_in[0];
  const int*   items = (const int*)d_in[1];
  const int*   rules = (const int*)d_in[2];
  const int*   adj   = (const int*)d_in[3];
  const float* ent   = (const float*)d_in[4];
  const float* rw    = (const float*)d_in[5];
  const float* W     = (const float*)d_in[6];
  const float* bias  = (const float*)d_in[7];
  float* outp = (float*)d_out;
  rgrec_kernel<<<BATCH, 256, 0, stream>>>(users, items, rules, adj, ent, rw, W, bias, outp);
}